// MPLPNodeLabel_61512521613939
// MI455X (gfx1250) — compile-verified
//
#include <hip/hip_runtime.h>
#include <hip/hip_bf16.h>

// N=100000, DIM=128, M=1600000, E=200000, HOP=2
// X layout in workspace: [N][9][128] f32, row pitch 1152 floats.

typedef float v2f __attribute__((ext_vector_type(2)));
typedef float v8f __attribute__((ext_vector_type(8)));

#define KPITCH 1152  // 9*128

// ---------------------------------------------------------------------------
// Phase A: build channels 0..2 = x_sig * {1, rsqrt(deg), rsqrt(1+log deg)},
// zero channels 3..8 (spmm accumulators).
// ---------------------------------------------------------------------------
__global__ void build_x_kernel(const float* __restrict__ x_sig,
                               const float* __restrict__ deg,
                               float* __restrict__ X, int N) {
    long tid = (long)blockIdx.x * blockDim.x + threadIdx.x;
    if (tid >= (long)N * 128) return;
    long i = tid >> 7;
    int  d = (int)(tid & 127);
    float s  = x_sig[tid];
    float dg = deg[i];
    float w1 = rsqrtf(dg);
    float w2 = rsqrtf(1.0f + logf(dg));
    float* p = X + i * (long)KPITCH;
    p[d]         = s;
    p[128 + d]   = s * w1;
    p[256 + d]   = s * w2;
#pragma unroll
    for (int k = 3; k < 9; ++k) p[k * 128 + d] = 0.0f;
}

// ---------------------------------------------------------------------------
// Phase B: scatter spmm. One wave per nnz: dst[kd..kd+2][*] += src[ks..ks+2][*]
// 384 floats per nnz, 12 coalesced elements per lane, f32 global atomics.
// ---------------------------------------------------------------------------
__global__ void spmm_scatter_kernel(const int* __restrict__ rows,
                                    const int* __restrict__ cols,
                                    float* __restrict__ X,
                                    int ks, int kd, long M) {
    long nz = (long)blockIdx.x * 8 + (threadIdx.x >> 5);
    if (nz >= M) return;
    int lane = threadIdx.x & 31;
    long r = rows[nz];
    long c = cols[nz];
    const float* src = X + c * (long)KPITCH + (long)ks * 128;
    float*       dst = X + r * (long)KPITCH + (long)kd * 128;
#pragma unroll
    for (int t = 0; t < 12; ++t) {
        int idx = t * 32 + lane;
        float v = src[idx];
        atomicAdd(&dst[idx], v);  // global_atomic_add_f32 (no-return)
    }
}

// Scale hop-1 accumulators by 1/deg.
__global__ void scale_hop1_kernel(const float* __restrict__ deg,
                                  float* __restrict__ X, int N) {
    long tid = (long)blockIdx.x * blockDim.x + threadIdx.x;
    if (tid >= (long)N * 128) return;
    long i = tid >> 7;
    int  d = (int)(tid & 127);
    float dr = 1.0f / deg[i];
    float* p = X + i * (long)KPITCH;
    p[384 + d] *= dr;
    p[512 + d] *= dr;
    p[640 + d] *= dr;
}

// Finalize hop-2: ch6..8 = (1/deg)*acc - ch0..2   (Chebyshev-style recurrence)
__global__ void finalize_hop2_kernel(const float* __restrict__ deg,
                                     float* __restrict__ X, int N) {
    long tid = (long)blockIdx.x * blockDim.x + threadIdx.x;
    if (tid >= (long)N * 128) return;
    long i = tid >> 7;
    int  d = (int)(tid & 127);
    float dr = 1.0f / deg[i];
    float* p = X + i * (long)KPITCH;
    p[768 + d]  = dr * p[768 + d]  - p[d];
    p[896 + d]  = dr * p[896 + d]  - p[128 + d];
    p[1024 + d] = dr * p[1024 + d] - p[256 + d];
}

// ---------------------------------------------------------------------------
// Phase C: per-edge Gram matrix via V_WMMA_F32_16X16X4_F32.
// One edge per wave32; 9x128 A/B padded into 16x128; K consumed in 32 steps
// of K=4. A frag: lane<16 -> row=lane, K={k0,k0+1}; lane>=16 -> K={k0+2,k0+3}.
// B frag mirrors with lane<->column.
//
// No masking needed for the padded rows: G[k][l] depends only on A-row k and
// B-col l, and we only read the k,l < 9 block of the 16x16 tile. Lanes with
// r >= 9 clamp to row 8 (finite real data, cache lines already fetched), so
// garbage lands exclusively in tile entries we never read. This keeps the
// inner loop branch-free: 2 x global_load_b64 (immediate offsets) + 1 WMMA,
// with EXEC all-ones throughout (WMMA requirement).
// ---------------------------------------------------------------------------
__global__ void __launch_bounds__(256)
edge_gram_kernel(const int* __restrict__ edge,
                 const float* __restrict__ X,
                 float* __restrict__ out, int E) {
    __shared__ float g[8][16 * 17];

    const int lane = threadIdx.x & 31;
    const int w    = threadIdx.x >> 5;
    const int half = lane >> 4;         // 0: K pair {0,1}; 1: K pair {2,3}
    const int r    = lane & 15;         // A row / B column within tile
    const int rc   = (r < 9) ? r : 8;   // clamp: pad rows duplicate row 8

    long e  = (long)blockIdx.x * 8 + w;
    long ec = (e < E) ? e : (long)(E - 1);   // keep all waves converged

    int u = edge[ec];            // edge[0][e]
    int v = edge[(long)E + ec];  // edge[1][e]

    const int base = rc * 128 + half * 2;
    const float* pu = X + (long)u * KPITCH + base;
    const float* pv = X + (long)v * KPITCH + base;

    v8f c = {0.0f, 0.0f, 0.0f, 0.0f, 0.0f, 0.0f, 0.0f, 0.0f};

#pragma unroll
    for (int t = 0; t < 32; ++t) {
        v2f a2 = *(const v2f*)(pu + 4 * t);   // global_load_b64, imm offset 16*t
        v2f b2 = *(const v2f*)(pv + 4 * t);
        c = __builtin_amdgcn_wmma_f32_16x16x4_f32(
                /*neg_a=*/false, a2, /*neg_b=*/false, b2,
                /*c_mod=*/(short)0, c, /*reuse_a=*/false, /*reuse_b=*/false);
    }

    // Stage 16x16 tile: VGPR j holds M=j (lanes 0-15) / M=j+8 (lanes 16-31),
    // N = lane&15. Padded (k>=9 or l>=9) entries are written but never read.
#pragma unroll
    for (int j = 0; j < 8; ++j) {
        g[w][(j + half * 8) * 17 + r] = c[j];
    }
    __syncthreads();

    // out[e][k*9+l] = G[k][l] + G[l][k], 81 entries, 3 per lane.
    if (e < E) {
#pragma unroll
        for (int s = 0; s < 3; ++s) {
            int idx = lane + 32 * s;
            if (idx < 81) {
                int k = idx / 9;
                int l = idx - 9 * k;
                out[e * 81 + idx] = g[w][k * 17 + l] + g[w][l * 17 + k];
            }
        }
    }
}

// ---------------------------------------------------------------------------
extern "C" void kernel_launch(void* const* d_in, const int* in_sizes, int n_in,
                              void* d_out, int out_size, void* d_ws, size_t ws_size,
                              hipStream_t stream) {
    const int*   edge    = (const int*)d_in[0];   // [2,E] int32
    const int*   adj_row = (const int*)d_in[1];   // [M]
    const int*   adj_col = (const int*)d_in[2];   // [M]
    const float* deg     = (const float*)d_in[3]; // [N,1]
    const float* x_sig   = (const float*)d_in[4]; // [N,128]
    float*       out     = (float*)d_out;         // [E,81]
    float*       X       = (float*)d_ws;          // [N,9,128] = 460.8 MB

    const int  N = in_sizes[3];
    const long M = in_sizes[1];
    const int  E = in_sizes[0] / 2;

    const int  threads = 256;
    const long nodeThreads = (long)N * 128;
    const int  gridA = (int)((nodeThreads + threads - 1) / threads);
    const int  gridS = (int)((M + 7) / 8);
    const int  gridE = (int)(((long)E + 7) / 8);

    build_x_kernel<<<gridA, threads, 0, stream>>>(x_sig, deg, X, N);
    spmm_scatter_kernel<<<gridS, threads, 0, stream>>>(adj_row, adj_col, X, 0, 3, M);
    scale_hop1_kernel<<<gridA, threads, 0, stream>>>(deg, X, N);
    spmm_scatter_kernel<<<gridS, threads, 0, stream>>>(adj_row, adj_col, X, 3, 6, M);
    finalize_hop2_kernel<<<gridA, threads, 0, stream>>>(deg, X, N);
    edge_gram_kernel<<<gridE, threads, 0, stream>>>(edge, X, out, E);
}